// GCN_15865609192043
// MI455X (gfx1250) — compile-verified
//
#include <hip/hip_runtime.h>
#include <hip/hip_bf16.h>

// fp32 WMMA fragments (wave32): A/B = 2 VGPRs, C/D = 8 VGPRs
typedef float v2f __attribute__((ext_vector_type(2)));
typedef float v8f __attribute__((ext_vector_type(8)));

#define WMMA_F32X4(a, b, c) \
  __builtin_amdgcn_wmma_f32_16x16x4_f32(false, (a), false, (b), (short)0, (c), false, false)

// ---------------------------------------------------------------------------
// lin1: out[N,8] = relu(x[N,128] @ W^T[128,8] + b)
// One wave per 16-row tile; cols padded 8->16 via clamped-address + mask-mul
// (no divergent loads: EXEC stays clean around the WMMA chain).
// ---------------------------------------------------------------------------
__global__ __launch_bounds__(128) void k_lin1(const float* __restrict__ x,
                                              const float* __restrict__ W,   // [8,128]
                                              const float* __restrict__ b,   // [8]
                                              float* __restrict__ out,       // [N,8]
                                              int nTiles) {
  const int tile = blockIdx.x * 4 + threadIdx.y;
  if (tile >= nTiles) return;                       // uniform per wave
  const int lane = threadIdx.x;
  const int row  = lane & 15;
  const int hi   = lane >> 4;
  const int m0   = tile << 4;

  const float* xr   = x + (m0 + row) * 128;
  const float  mask = (row < 8) ? 1.0f : 0.0f;      // zero-pad cols 8..15
  const float* wr   = W + (row & 7) * 128;          // always in-bounds

  v8f c = {};
#pragma unroll
  for (int k0 = 0; k0 < 128; k0 += 4) {
    const int ka = k0 + 2 * hi;
    v2f a, bb;
    a.x  = xr[ka];
    a.y  = xr[ka + 1];
    bb.x = wr[ka]     * mask;
    bb.y = wr[ka + 1] * mask;
    c = WMMA_F32X4(a, bb, c);
  }
  if (row < 8) {
    const float bias = b[row];
#pragma unroll
    for (int v = 0; v < 8; ++v) {
      const int m = m0 + v + 8 * hi;               // D: M = v + 8*hi, N = row
      float val = c[v] + bias;
      out[m * 8 + row] = val > 0.f ? val : 0.f;
    }
  }
}

// ---------------------------------------------------------------------------
// GraphConv dense part: out[N,32] = (agg @ Wrel^T + brel + h @ Wroot^T), opt relu
// K compile-time (8 or 32) -> fully unrolled WMMA chain, b64-mergeable loads.
// Grid.y = 2 col tiles (Cout = 32).
// ---------------------------------------------------------------------------
template <int K, bool RELU>
__global__ __launch_bounds__(128) void k_convlin(const float* __restrict__ agg,
                                                 const float* __restrict__ h,
                                                 const float* __restrict__ Wrel,
                                                 const float* __restrict__ brel,
                                                 const float* __restrict__ Wroot,
                                                 float* __restrict__ out,
                                                 int nTiles) {
  const int tile = blockIdx.x * 4 + threadIdx.y;
  if (tile >= nTiles) return;
  const int n0   = blockIdx.y << 4;
  const int lane = threadIdx.x;
  const int row  = lane & 15;
  const int hi   = lane >> 4;
  const int m0   = tile << 4;

  v8f c = {};
  {
    const float* ar = agg  + (m0 + row) * K;
    const float* br = Wrel + (n0 + row) * K;
#pragma unroll
    for (int k0 = 0; k0 < K; k0 += 4) {
      const int ka = k0 + 2 * hi;
      v2f a, bb;
      a.x = ar[ka];  a.y = ar[ka + 1];
      bb.x = br[ka]; bb.y = br[ka + 1];
      c = WMMA_F32X4(a, bb, c);
    }
  }
  {
    const float* hr = h     + (m0 + row) * K;
    const float* br = Wroot + (n0 + row) * K;
#pragma unroll
    for (int k0 = 0; k0 < K; k0 += 4) {
      const int ka = k0 + 2 * hi;
      v2f a, bb;
      a.x = hr[ka];  a.y = hr[ka + 1];
      bb.x = br[ka]; bb.y = br[ka + 1];
      c = WMMA_F32X4(a, bb, c);
    }
  }
  const float bias = brel[n0 + row];
#pragma unroll
  for (int v = 0; v < 8; ++v) {
    const int m = m0 + v + 8 * hi;
    float val = c[v] + bias;
    if (RELU) val = val > 0.f ? val : 0.f;
    out[m * 32 + n0 + row] = val;
  }
}

// ---------------------------------------------------------------------------
// Edge messages: agg[dst] += w * h[src], C channels as float4 chunks.
// Dominant kernel: ~E*C*8 bytes of gather + atomic traffic.
// ---------------------------------------------------------------------------
__global__ void k_edge(const float* __restrict__ h,
                       const int* __restrict__ src, const int* __restrict__ dst,
                       const float* __restrict__ ew,
                       float* __restrict__ agg, int E, int C4) {
  const int total = E * C4;
  for (int i = blockIdx.x * blockDim.x + threadIdx.x; i < total;
       i += gridDim.x * blockDim.x) {
    const int e  = i / C4;
    const int c4 = i - e * C4;
    const int s  = src[e];
    const int d  = dst[e];
    const float w = ew[e];
    const float4 v = ((const float4*)(h + s * (C4 * 4)))[c4];
    float* ap = agg + d * (C4 * 4) + c4 * 4;
    atomicAdd(ap + 0, w * v.x);
    atomicAdd(ap + 1, w * v.y);
    atomicAdd(ap + 2, w * v.z);
    atomicAdd(ap + 3, w * v.w);
  }
}

// ---------------------------------------------------------------------------
// Mean-pool accumulation: sums[batch[n],c] += h[n,c]; counts[batch[n]] += 1
// ---------------------------------------------------------------------------
__global__ void k_pool(const float* __restrict__ h, const int* __restrict__ batch,
                       float* __restrict__ sums, float* __restrict__ counts, int N) {
  const int total = N * 32;
  for (int i = blockIdx.x * blockDim.x + threadIdx.x; i < total;
       i += gridDim.x * blockDim.x) {
    const int n = i >> 5;
    const int c = i & 31;
    const int g = batch[n];
    atomicAdd(&sums[g * 32 + c], h[i]);
    if (c == 0) atomicAdd(&counts[g], 1.0f);
  }
}

__global__ void k_zero4(float4* __restrict__ p, int n4) {
  const float4 z = {0.f, 0.f, 0.f, 0.f};
  for (int i = blockIdx.x * blockDim.x + threadIdx.x; i < n4;
       i += gridDim.x * blockDim.x) p[i] = z;
}

// ---------------------------------------------------------------------------
// Fused head: pooled = sums/max(counts,1); z = relu(pooled @ Wa^T + ba) via WMMA,
// staged in LDS; out = z @ Wb^T + bb via WMMA (10 cols padded to 16).
// All pad/guard loads use clamped addresses + mask-mul (no EXEC churn).
// ---------------------------------------------------------------------------
__global__ __launch_bounds__(128) void k_lin2(const float* __restrict__ sums,
                                              const float* __restrict__ counts,
                                              const float* __restrict__ Wa,   // [16,32]
                                              const float* __restrict__ ba,   // [16]
                                              const float* __restrict__ Wb,   // [10,16]
                                              const float* __restrict__ bb2,  // [10]
                                              float* __restrict__ out,        // [G,10]
                                              int G, int nTiles) {
  __shared__ float zbuf[4][16 * 17];
  const int tile = blockIdx.x * 4 + threadIdx.y;
  const bool active = tile < nTiles;                // uniform per wave
  const int lane = threadIdx.x;
  const int row  = lane & 15;
  const int hi   = lane >> 4;
  const int g0   = tile << 4;
  float* zt = zbuf[threadIdx.y];

  if (active) {
    const int   gr   = g0 + row;
    const int   grc  = gr < G ? gr : (G - 1);       // clamped row
    // invc==0 for padded rows -> A fragment is zero without branching
    const float invc = (gr < G) ? 1.0f / fmaxf(counts[grc], 1.0f) : 0.f;
    const float* sr  = sums + grc * 32;
    const float* war = Wa + row * 32;               // all 16 out-cols valid
    v8f c = {};
#pragma unroll
    for (int k0 = 0; k0 < 32; k0 += 4) {
      const int ka = k0 + 2 * hi;
      v2f a, b;
      a.x = sr[ka]     * invc;
      a.y = sr[ka + 1] * invc;
      b.x = war[ka];
      b.y = war[ka + 1];
      c = WMMA_F32X4(a, b, c);
    }
    const float biasA = ba[row];
#pragma unroll
    for (int v = 0; v < 8; ++v) {
      float val = c[v] + biasA;
      zt[(v + 8 * hi) * 17 + row] = val > 0.f ? val : 0.f;
    }
  }
  __syncthreads();
  if (active) {
    const float  maskB = (row < 10) ? 1.0f : 0.0f;  // zero-pad cols 10..15
    const float* wbr   = Wb + (row < 10 ? row : 0) * 16;
    const float* zr    = zt + row * 17;
    v8f c2 = {};
#pragma unroll
    for (int k0 = 0; k0 < 16; k0 += 4) {
      const int ka = k0 + 2 * hi;
      v2f a, b;
      a.x = zr[ka];
      a.y = zr[ka + 1];
      b.x = wbr[ka]     * maskB;
      b.y = wbr[ka + 1] * maskB;
      c2 = WMMA_F32X4(a, b, c2);
    }
    if (row < 10) {
      const float biasB = bb2[row];
#pragma unroll
      for (int v = 0; v < 8; ++v) {
        const int g = g0 + v + 8 * hi;
        if (g < G) out[g * 10 + row] = c2[v] + biasB;
      }
    }
  }
}

// ---------------------------------------------------------------------------
extern "C" void kernel_launch(void* const* d_in, const int* in_sizes, int n_in,
                              void* d_out, int out_size, void* d_ws, size_t ws_size,
                              hipStream_t stream) {
  (void)n_in; (void)ws_size;
  const float* x      = (const float*)d_in[0];
  const int*   ei     = (const int*)d_in[1];     // [2,E]: src then dst
  const float* ew     = (const float*)d_in[2];
  const int*   batch  = (const int*)d_in[3];
  const float* W_lin1 = (const float*)d_in[4];
  const float* b_lin1 = (const float*)d_in[5];
  const float* W1_rel = (const float*)d_in[6];
  const float* b1_rel = (const float*)d_in[7];
  const float* W1_root= (const float*)d_in[8];
  const float* W2_rel = (const float*)d_in[9];
  const float* b2_rel = (const float*)d_in[10];
  const float* W2_root= (const float*)d_in[11];
  const float* W3_rel = (const float*)d_in[12];
  const float* b3_rel = (const float*)d_in[13];
  const float* W3_root= (const float*)d_in[14];
  const float* W_l2a  = (const float*)d_in[15];
  const float* b_l2a  = (const float*)d_in[16];
  const float* W_l2b  = (const float*)d_in[17];
  const float* b_l2b  = (const float*)d_in[18];
  float* out = (float*)d_out;

  const int N = in_sizes[0] / 128;   // 100000 (multiple of 16)
  const int E = in_sizes[2];         // 1600000
  const int G = out_size / 10;       // 1000
  const int* srcIdx = ei;
  const int* dstIdx = ei + E;

  // workspace layout (floats): ~41.8 MB total
  float* ws     = (float*)d_ws;
  float* h1     = ws;                 // N*8
  float* agg    = h1  + N * 8;        // N*32
  float* hA     = agg + N * 32;       // N*32
  float* hB     = hA  + N * 32;       // N*32
  float* sums   = hB  + N * 32;       // G*32
  float* counts = sums + G * 32;      // G   (contiguous after sums)

  const int rowTiles = N / 16;        // 6250
  const dim3 wblk(32, 4);
  const dim3 gemmGrid((rowTiles + 3) / 4, 2);

  // lin1 + relu
  k_lin1<<<dim3((rowTiles + 3) / 4), wblk, 0, stream>>>(x, W_lin1, b_lin1, h1, rowTiles);

  // conv1 (C=8 -> 32) + relu
  k_zero4<<<1024, 256, 0, stream>>>((float4*)agg, N * 8 / 4);
  k_edge<<<4096, 256, 0, stream>>>(h1, srcIdx, dstIdx, ew, agg, E, 2);
  k_convlin<8, true><<<gemmGrid, wblk, 0, stream>>>(agg, h1, W1_rel, b1_rel, W1_root,
                                                    hA, rowTiles);
  // conv2 (32 -> 32) + relu
  k_zero4<<<1024, 256, 0, stream>>>((float4*)agg, N * 32 / 4);
  k_edge<<<4096, 256, 0, stream>>>(hA, srcIdx, dstIdx, ew, agg, E, 8);
  k_convlin<32, true><<<gemmGrid, wblk, 0, stream>>>(agg, hA, W2_rel, b2_rel, W2_root,
                                                     hB, rowTiles);
  // conv3 (32 -> 32), no relu
  k_zero4<<<1024, 256, 0, stream>>>((float4*)agg, N * 32 / 4);
  k_edge<<<4096, 256, 0, stream>>>(hB, srcIdx, dstIdx, ew, agg, E, 8);
  k_convlin<32, false><<<gemmGrid, wblk, 0, stream>>>(agg, hB, W3_rel, b3_rel, W3_root,
                                                      hA, rowTiles);

  // global mean pool  (sums + counts are contiguous: zero both)
  k_zero4<<<64, 256, 0, stream>>>((float4*)sums, (G * 32 + G) / 4);
  k_pool<<<8192, 256, 0, stream>>>(hA, batch, sums, counts, N);

  // fused lin2a + relu + lin2b
  const int gTiles = (G + 15) / 16;   // 63
  k_lin2<<<dim3((gTiles + 3) / 4), wblk, 0, stream>>>(sums, counts, W_l2a, b_l2a,
                                                      W_l2b, b_l2b, out, G, gTiles);
}